// Embedding_6940667150787
// MI455X (gfx1250) — compile-verified
//
#include <hip/hip_runtime.h>

typedef __attribute__((ext_vector_type(4))) float v4f;
typedef int v4i_g __attribute__((vector_size(16)));   // matches builtin param type

#define DEPTH 512
#define NPAIR 255
#define STEPS 16
#define ROWS_PER_BLOCK (2 * STEPS)   // 2 rows per pipeline step

#if __has_builtin(__builtin_amdgcn_global_load_async_to_lds_b128) && \
    __has_builtin(__builtin_amdgcn_s_wait_asynccnt)
#define HAVE_ASYNC_LDS 1
#else
#define HAVE_ASYNC_LDS 0
#endif

// log2(10000) / 256  (inverse frequency exponent step)
#define L2_10000_OVER_256 (13.287712379549449f / 256.0f)

__global__ __launch_bounds__(256)
void emb_pe_kernel(const int* __restrict__ idx,
                   const float* __restrict__ wl,
                   float* __restrict__ out,
                   int nrows) {
  const int tid  = threadIdx.x;
  const int lane = tid & 127;    // which 16-byte column group of the row
  const int rsub = tid >> 7;     // which of the 2 rows this step
  const int col  = lane << 2;    // starting column (4 floats per thread)
  const int j0   = lane << 1;    // sin/cos pair indices j0, j0+1
  const int j1   = j0 + 1;

  // inv_freq(j) = 10000^(-j/256) = exp2(-j * log2(10000)/256)
  const float f0  = exp2f(-(float)j0 * L2_10000_OVER_256);
  const float f1  = exp2f(-(float)j1 * L2_10000_OVER_256);
  const bool  pe1 = (j1 < NPAIR);          // pair 255 (cols 510/511): PE == 0

  const int base = blockIdx.x * ROWS_PER_BLOCK;
  const int last = nrows - 1;

#if HAVE_ASYNC_LDS
  __shared__ float lds[2][256 * 4];        // 8 KB double buffer, 16 B per thread
#endif

  // clamped row used only for safe idx/table reads; stores are guarded exactly
  int r0 = base + rsub;
  int rc0 = r0 < nrows ? r0 : last;

#if HAVE_ASYNC_LDS
  {
    int t0 = idx[rc0];
    __builtin_amdgcn_global_load_async_to_lds_b128(
        (v4i_g*)(wl + (size_t)t0 * DEPTH + col),
        (v4i_g*)(&lds[0][tid * 4]), 0, 0);
  }
#endif

#pragma unroll 4
  for (int k = 0; k < STEPS; ++k) {
    const int row = base + 2 * k + rsub;

#if HAVE_ASYNC_LDS
    if (k + 1 < STEPS) {
      int rn = row + 2;
      int rcn = rn < nrows ? rn : last;
      int tn = idx[rcn];
      __builtin_amdgcn_global_load_async_to_lds_b128(
          (v4i_g*)(wl + (size_t)tn * DEPTH + col),
          (v4i_g*)(&lds[(k + 1) & 1][tid * 4]), 0, 0);
      __builtin_amdgcn_s_wait_asynccnt(1);   // step-k copy complete
    } else {
      __builtin_amdgcn_s_wait_asynccnt(0);
    }
    v4f e = *(const v4f*)&lds[k & 1][tid * 4];
#else
    int rck = row < nrows ? row : last;
    int t = idx[rck];
    v4f e = *(const v4f*)(wl + (size_t)t * DEPTH + col);
#endif

    const float fr = (float)row;
    float s0, c0, s1, c1;
    __sincosf(fr * f0, &s0, &c0);
    __sincosf(fr * f1, &s1, &c1);

    v4f o;
    o.x = e.x + s0;
    o.y = e.y + c0;
    o.z = e.z + (pe1 ? s1 : 0.0f);
    o.w = e.w + (pe1 ? c1 : 0.0f);

    if (row < nrows) {
      // streaming 256 MB output: bypass L2 retention (keep L2 for the table)
      __builtin_nontemporal_store(o, (v4f*)(out + (size_t)row * DEPTH + col));
    }
  }
}

extern "C" void kernel_launch(void* const* d_in, const int* in_sizes, int n_in,
                              void* d_out, int out_size, void* d_ws, size_t ws_size,
                              hipStream_t stream) {
  const int*   idx = (const int*)d_in[0];
  const float* wl  = (const float*)d_in[1];
  float*       out = (float*)d_out;
  const int nrows  = in_sizes[0];          // 131072

  const int grid = (nrows + ROWS_PER_BLOCK - 1) / ROWS_PER_BLOCK;   // 4096
  emb_pe_kernel<<<grid, 256, 0, stream>>>(idx, wl, out, nrows);
}